// Executor_48515950576547
// MI455X (gfx1250) — compile-verified
//
#include <hip/hip_runtime.h>
#include <cstdint>
#include <cstddef>

// ---------------------------------------------------------------------------
// Fused gather + score(WMMA f32 16x16x4) + tanh-weighted recombine + residual
// B=8 S=2048 K=16 D=512, POOL=500000.  One wave32 per token.
// Memory-bound: ~600MB/iter -> ~26us floor @ 23.3 TB/s HBM.
// Gather path: global_load_async_to_lds_b128 (ASYNCcnt DMA, no VGPR round-trip).
// Score path : v_wmma_f32_16x16x4_f32, A = gathered 16xD tile, B = x bcast.
// ---------------------------------------------------------------------------

typedef __attribute__((ext_vector_type(2))) float v2f;
typedef __attribute__((ext_vector_type(8))) float v8f;
typedef __attribute__((ext_vector_type(4))) int   v4i;

#define AS1 __attribute__((address_space(1)))
#define AS3 __attribute__((address_space(3)))

#ifndef __has_builtin
#define __has_builtin(x) 0
#endif

#if __has_builtin(__builtin_amdgcn_global_load_async_to_lds_b128)
#define USE_ASYNC_LDS 1
#else
#define USE_ASYNC_LDS 0
#endif

constexpr int kBatch  = 8;
constexpr int kSeq    = 2048;
constexpr int kK      = 16;
constexpr int kD      = 512;
constexpr int kTokens = kBatch * kSeq;                 // 16384

constexpr int SEL_STRIDE = kD + 4;                     // 516 floats: bank-conflict pad
constexpr int WAVE_FLOATS = kK * SEL_STRIDE + kD + 16 + 16; // sel + xbuf + pbuf + tbuf = 8800
constexpr int WAVES_PER_BLOCK = 4;

__global__ __launch_bounds__(WAVES_PER_BLOCK * 32)
void moe_gather_wmma_kernel(const float* __restrict__ x,
                            const int*   __restrict__ indices,
                            const float* __restrict__ weights,
                            const float* __restrict__ pool,
                            float*       __restrict__ out)
{
    extern __shared__ float smem[];
    const int lane  = threadIdx.x & 31;
    const int wave  = threadIdx.x >> 5;
    const int token = blockIdx.x * WAVES_PER_BLOCK + wave;

    float* base = smem + wave * WAVE_FLOATS;
    float* sel  = base;                       // 16 rows x 516 floats
    float* xbuf = base + kK * SEL_STRIDE;     // 512 floats
    float* pbuf = xbuf + kD;                  // 16 raw scores
    float* tbuf = pbuf + 16;                  // 16 tanh*weight values

    // ---- stage x[token] into LDS (coalesced float4) ----
    const float* xrow = x + (size_t)token * kD;
    #pragma unroll
    for (int c = 0; c < kD / 128; ++c) {
        const int f = (c * 32 + lane) * 4;
        *(float4*)(xbuf + f) = *(const float4*)(xrow + f);
    }

    // ---- load the 16 gather indices (uniform within the wave) ----
    int idxv[kK];
    #pragma unroll
    for (int k = 0; k < kK; ++k) idxv[k] = indices[(size_t)token * kK + k];

    // ---- gather 16 pool rows (2KB each) into LDS, coalesced b128 DMA ----
    #pragma unroll
    for (int k = 0; k < kK; ++k) {
        const float* src = pool + (size_t)idxv[k] * kD;
        float*       dst = sel + k * SEL_STRIDE;
        #pragma unroll
        for (int c = 0; c < kD / 128; ++c) {
            const int f = (c * 32 + lane) * 4;
#if USE_ASYNC_LDS
            __builtin_amdgcn_global_load_async_to_lds_b128(
                (AS1 v4i*)(src + f), (AS3 v4i*)(dst + f), 0, 0);
#else
            *(float4*)(dst + f) = *(const float4*)(src + f);
#endif
        }
    }
#if USE_ASYNC_LDS
  #if __has_builtin(__builtin_amdgcn_s_wait_asynccnt)
    __builtin_amdgcn_s_wait_asynccnt(0);
  #else
    asm volatile("s_wait_asynccnt 0" ::: "memory");
  #endif
#endif

    // ---- scores: C(16x16) += A(16x4 tile slice) * B(4x16 = x chunk bcast) ----
    // A layout (ISA 7.12.2, 32-bit 16x4): lanes 0-15 hold K=0,1; lanes 16-31 K=2,3.
    // B symmetric: lanes 0-15 hold K=0,1 (col = lane), lanes 16-31 K=2,3.
    // Replicating x across all columns makes every C column equal the score vector.
    const int row  = lane & 15;
    const int half = lane >> 4;
    const float* arow = sel + row * SEL_STRIDE + 2 * half;
    const float* brow = xbuf + 2 * half;

    v8f acc = {0.f, 0.f, 0.f, 0.f, 0.f, 0.f, 0.f, 0.f};
    #pragma unroll 8
    for (int d0 = 0; d0 < kD; d0 += 4) {
        v2f a = *(const v2f*)(arow + d0);
        v2f b = *(const v2f*)(brow + d0);
        acc = __builtin_amdgcn_wmma_f32_16x16x4_f32(
            false, a, false, b, (short)0, acc, false, false);
    }

    // ---- extract scores (C layout: lane0 VGPRv = C[v,0], lane16 = C[v+8,0]) ----
    // Lanes 0 and 16 each pack their 8 accumulator regs as two b128 stores.
    if ((lane & 15) == 0) {
        float* p = pbuf + half * 8;
        *(float4*)(p + 0) = make_float4(acc[0], acc[1], acc[2], acc[3]);
        *(float4*)(p + 4) = make_float4(acc[4], acc[5], acc[6], acc[7]);
    }
    // Branch-free tanh*weight: each m handled twice with identical values.
    {
        const int   mk = lane & 15;
        const float p  = pbuf[mk];               // DS in-order w.r.t. stores above
        const float w  = weights[(size_t)token * kK + mk];
        tbuf[mk] = tanhf(p) * w;
    }

    // ---- recombine: out[d] = sum_m t[m]*sel[m][d] + x[d], lane-contiguous ----
    float4 tv4[4];
    #pragma unroll
    for (int q = 0; q < 4; ++q) tv4[q] = *(const float4*)(tbuf + q * 4);
    float tval[kK];
    #pragma unroll
    for (int q = 0; q < 4; ++q) {
        tval[q * 4 + 0] = tv4[q].x; tval[q * 4 + 1] = tv4[q].y;
        tval[q * 4 + 2] = tv4[q].z; tval[q * 4 + 3] = tv4[q].w;
    }

    float4 accv[4];
    #pragma unroll
    for (int j = 0; j < 4; ++j) {
        const int f = j * 128 + lane * 4;
        accv[j] = *(const float4*)(xbuf + f);      // residual seed
    }
    #pragma unroll
    for (int m = 0; m < kK; ++m) {
        const float tm = tval[m];
        const float* srow = sel + m * SEL_STRIDE;
        #pragma unroll
        for (int j = 0; j < 4; ++j) {
            const int f = j * 128 + lane * 4;
            const float4 s = *(const float4*)(srow + f);
            accv[j].x += tm * s.x;
            accv[j].y += tm * s.y;
            accv[j].z += tm * s.z;
            accv[j].w += tm * s.w;
        }
    }

    float* orow = out + (size_t)token * kD;
    #pragma unroll
    for (int j = 0; j < 4; ++j) {
        *(float4*)(orow + j * 128 + lane * 4) = accv[j];
    }
}

extern "C" void kernel_launch(void* const* d_in, const int* in_sizes, int n_in,
                              void* d_out, int out_size, void* d_ws, size_t ws_size,
                              hipStream_t stream) {
    const float* x       = (const float*)d_in[0];
    const int*   indices = (const int*)d_in[1];
    const float* weights = (const float*)d_in[2];
    const float* pool    = (const float*)d_in[3];
    float*       out     = (float*)d_out;

    dim3 grid(kTokens / WAVES_PER_BLOCK);
    dim3 block(WAVES_PER_BLOCK * 32);
    const size_t shmem = (size_t)WAVES_PER_BLOCK * WAVE_FLOATS * sizeof(float);

    hipLaunchKernelGGL(moe_gather_wmma_kernel, grid, block, shmem, stream,
                       x, indices, weights, pool, out);
}